// AxialBlock_49598282334314
// MI455X (gfx1250) — compile-verified
//
#include <hip/hip_runtime.h>
#include <hip/hip_bf16.h>
#include <math.h>

typedef __attribute__((ext_vector_type(2))) float v2f;
typedef __attribute__((ext_vector_type(8))) float v8f;

#define BSZ   16
#define CIN   128
#define HWD   56      // spatial H=W=56
#define CW    64      // width_c
#define L     56      // attention sequence length
#define NPOS  3136    // 56*56
#define NGRP  8

// ---- LDS layout (floats) for the fused attention kernel ----
// NOTE: attn kernel has no static __shared__, so dynamic LDS starts at byte 0;
// these float offsets are also the literal LDS offsets used by the async loads.
#define REL_OFF   0                 // 16 x 112  (rel table, padded 111->112)
#define REL_SZ    (16*112)
#define LX_OFF    (REL_OFF+REL_SZ)  // 64 x 64   (input column; reused for sve)
#define LX_SZ     (64*64)
#define LQKV_OFF  (LX_OFF+LX_SZ)    // 144 x 64  (BN'd qkv, rows padded 128->144)
#define LQKV_SZ   (144*64)
#define SIM_OFF   (LQKV_OFF+LQKV_SZ)// 8 x 56 x 57
#define SIMROW    57
#define SIMGRP    (L*SIMROW)
#define SIM_SZ    (NGRP*SIMGRP)
#define SMEM_FLOATS (SIM_OFF+SIM_SZ)
#define SMEM_BYTES  (SMEM_FLOATS*4)

__device__ inline v8f wmma_f32(v2f a, v2f b, v8f c) {
  // V_WMMA_F32_16X16X4_F32 : D(16x16,f32) = A(16x4,f32) x B(4x16,f32) + C
  return __builtin_amdgcn_wmma_f32_16x16x4_f32(
      false, a, false, b, (short)0, c, false, false);
}

// CDNA5 async copy global->LDS (GV mode, per-lane addresses, ASYNCcnt-tracked)
__device__ inline void async_ld_f32(unsigned int lds_byte_off, const float* g) {
  asm volatile("global_load_async_to_lds_b32 %0, %1, off"
               :: "v"(lds_byte_off), "v"((unsigned long long)(uintptr_t)g)
               : "memory");
}
__device__ inline void wait_async0() {
  asm volatile("s_wait_asynccnt 0x0" ::: "memory");
}

// BN params stored as [gamma | beta | mean | var], each C long.
// Returns (scale, shift): y = x*scale + shift
__device__ inline float2 bn_coef(const float* __restrict__ p, int C, int ch) {
  float g = p[ch], bb = p[C + ch], m = p[2 * C + ch], v = p[3 * C + ch];
  float s = g * rsqrtf(v + 1e-5f);
  return make_float2(s, bb - m * s);
}

// ---------------------------------------------------------------------------
// conv_down: y1[b,o,pos] = relu(bn1(sum_c W[o,c]*x[b,c,pos])), o<64, c<128
// weights staged in LDS; one wave per 16-wide position tile; 32 wmma k-steps
// ---------------------------------------------------------------------------
__global__ __launch_bounds__(256)
void conv_down_kernel(const float* __restrict__ x, const float* __restrict__ w,
                      const float* __restrict__ bnp, float* __restrict__ y) {
  __shared__ float lw[CW * CIN];  // 32 KB
  for (int idx = threadIdx.x; idx < CW * CIN; idx += 256) lw[idx] = w[idx];
  __syncthreads();

  int gw = blockIdx.x * 8 + (threadIdx.x >> 5);
  int b = gw / 196, nt = gw - b * 196;
  int lane = threadIdx.x & 31, l16 = lane & 15, lh = lane >> 4;
  int pos = nt * 16 + l16;
  const float* xb = x + (size_t)b * CIN * NPOS;
  const v8f vz = {0, 0, 0, 0, 0, 0, 0, 0};
  v8f acc[4] = {vz, vz, vz, vz};
#pragma unroll
  for (int kk = 0; kk < 32; ++kk) {
    int kb = kk * 4 + 2 * lh;
    if (kb + 32 < CIN)  // prefetch 8 k-steps ahead (global_prefetch_b8)
      __builtin_prefetch(xb + (size_t)(kb + 32) * NPOS + pos, 0, 1);
    v2f bf;
    bf.x = xb[(size_t)kb * NPOS + pos];
    bf.y = xb[(size_t)(kb + 1) * NPOS + pos];
#pragma unroll
    for (int mt = 0; mt < 4; ++mt) {
      v2f af;
      af.x = lw[(mt * 16 + l16) * CIN + kb];
      af.y = lw[(mt * 16 + l16) * CIN + kb + 1];
      acc[mt] = wmma_f32(af, bf, acc[mt]);
    }
  }
#pragma unroll
  for (int mt = 0; mt < 4; ++mt)
#pragma unroll
    for (int r = 0; r < 8; ++r) {
      int o = mt * 16 + r + 8 * lh;
      float2 s = bn_coef(bnp, CW, o);
      float v = acc[mt][r] * s.x + s.y;
      y[((size_t)b * CW + o) * NPOS + pos] = fmaxf(v, 0.f);
    }
}

// ---------------------------------------------------------------------------
// conv_up: out[b,o,pos] = relu(bn2(sum_c W[o,c]*relu(y3[b,c,pos])) + x[b,o,pos])
// o<128 (8 M-tiles), c<64 (16 k-steps); weights staged in LDS
// ---------------------------------------------------------------------------
__global__ __launch_bounds__(256)
void conv_up_kernel(const float* __restrict__ y3, const float* __restrict__ xres,
                    const float* __restrict__ w, const float* __restrict__ bnp,
                    float* __restrict__ out) {
  __shared__ float lw[CIN * CW];  // 32 KB
  for (int idx = threadIdx.x; idx < CIN * CW; idx += 256) lw[idx] = w[idx];
  __syncthreads();

  int gw = blockIdx.x * 8 + (threadIdx.x >> 5);
  int b = gw / 196, nt = gw - b * 196;
  int lane = threadIdx.x & 31, l16 = lane & 15, lh = lane >> 4;
  int pos = nt * 16 + l16;
  const float* yb = y3 + (size_t)b * CW * NPOS;
  const v8f vz = {0, 0, 0, 0, 0, 0, 0, 0};
  v8f acc[8] = {vz, vz, vz, vz, vz, vz, vz, vz};
#pragma unroll
  for (int kk = 0; kk < 16; ++kk) {
    int kb = kk * 4 + 2 * lh;
    if (kb + 32 < CW)
      __builtin_prefetch(yb + (size_t)(kb + 32) * NPOS + pos, 0, 1);
    v2f bf;
    bf.x = fmaxf(yb[(size_t)kb * NPOS + pos], 0.f);
    bf.y = fmaxf(yb[(size_t)(kb + 1) * NPOS + pos], 0.f);
#pragma unroll
    for (int mt = 0; mt < 8; ++mt) {
      v2f af;
      af.x = lw[(mt * 16 + l16) * CW + kb];
      af.y = lw[(mt * 16 + l16) * CW + kb + 1];
      acc[mt] = wmma_f32(af, bf, acc[mt]);
    }
  }
#pragma unroll
  for (int mt = 0; mt < 8; ++mt)
#pragma unroll
    for (int r = 0; r < 8; ++r) {
      int o = mt * 16 + r + 8 * lh;
      float2 s = bn_coef(bnp, CIN, o);
      float v = acc[mt][r] * s.x + s.y + xres[((size_t)b * CIN + o) * NPOS + pos];
      out[((size_t)b * CIN + o) * NPOS + pos] = fmaxf(v, 0.f);
    }
}

// ---------------------------------------------------------------------------
// Fused axial attention: one block per (batch, row). 8 waves, wave g = group g.
// WIDTH=false: sequence along H (input col x[b,c,:,r]); true: along W.
// ---------------------------------------------------------------------------
template <bool WIDTH>
__global__ __launch_bounds__(256)
void attn_kernel(const float* __restrict__ xin, const float* __restrict__ qkvw,
                 const float* __restrict__ bnqkv, const float* __restrict__ bnsim,
                 const float* __restrict__ bnout, const float* __restrict__ rel,
                 float* __restrict__ out) {
  extern __shared__ float smem[];
  float* lrel = smem + REL_OFF;
  float* lx = smem + LX_OFF;
  float* lqkv = smem + LQKV_OFF;
  float* lsim = smem + SIM_OFF;

  const int b = blockIdx.x / HWD;
  const int row = blockIdx.x % HWD;
  const int tid = threadIdx.x;
  const int wave = tid >> 5, lane = tid & 31, l16 = lane & 15, lh = lane >> 4;

  // ---- Phase 1: async-stage input column to LDS; stage rel; zero pads ----
  for (int idx = tid; idx < 64 * 64; idx += 256) {
    int c = idx >> 6, l = idx & 63;
    if (l < L) {
      const float* gp = WIDTH
          ? xin + (((size_t)b * CW + c) * HWD + row) * HWD + l
          : xin + (((size_t)b * CW + c) * HWD + l) * HWD + row;
      async_ld_f32((unsigned int)((LX_OFF + c * 64 + l) * 4), gp);
    } else {
      lx[c * 64 + l] = 0.f;
    }
  }
  for (int idx = tid; idx < 16 * 111; idx += 256) {
    int rr = idx / 111, cc = idx - rr * 111;
    lrel[rr * 112 + cc] = rel[idx];
  }
  for (int idx = tid; idx < 16; idx += 256) lrel[idx * 112 + 111] = 0.f;
  for (int idx = tid; idx < 16 * 64; idx += 256) lqkv[128 * 64 + idx] = 0.f;
  wait_async0();
  __syncthreads();

  // ---- Phase 2: qkv = Wqkv(128x64) @ X(64x56), BN, into LDS ----
  {
    v2f a[16];
#pragma unroll
    for (int kk = 0; kk < 16; ++kk) {
      int kb = kk * 4 + 2 * lh;
      a[kk].x = qkvw[(wave * 16 + l16) * CW + kb];
      a[kk].y = qkvw[(wave * 16 + l16) * CW + kb + 1];
    }
    const v8f vz = {0, 0, 0, 0, 0, 0, 0, 0};
    for (int nt = 0; nt < 4; ++nt) {
      v8f acc = vz;
#pragma unroll
      for (int kk = 0; kk < 16; ++kk) {
        int kb = kk * 4 + 2 * lh;
        v2f bf;
        bf.x = lx[kb * 64 + nt * 16 + l16];
        bf.y = lx[(kb + 1) * 64 + nt * 16 + l16];
        acc = wmma_f32(a[kk], bf, acc);
      }
#pragma unroll
      for (int r = 0; r < 8; ++r) {
        int o = wave * 16 + r + 8 * lh;
        float2 s = bn_coef(bnqkv, 2 * CW, o);
        lqkv[o * 64 + nt * 16 + l16] = acc[r] * s.x + s.y;
      }
    }
  }
  __syncthreads();  // after this, lx region is dead -> reused for sve results

  // ---- Phase 3 (per wave = per group g): qk^T, qr/kr, softmax, sve, sv ----
  const int g = wave;
  const float2 sQK = bn_coef(bnsim, 3 * NGRP, g);
  const float2 sQR = bn_coef(bnsim, 3 * NGRP, NGRP + g);
  const float2 sKR = bn_coef(bnsim, 3 * NGRP, 2 * NGRP + g);
  const float* qrows = lqkv + (g * 16) * 64;      // 4 rows: q
  const float* krows = lqkv + (g * 16 + 4) * 64;  // 4 rows: k
  const float* vrows = lqkv + (g * 16 + 8) * 64;  // 8 rows: v (pad-safe to 16)
  float* gsim = lsim + g * SIMGRP;
  float* gsve = lx + g * (8 * 64);                // this group's sve buffer
  const v8f vz = {0, 0, 0, 0, 0, 0, 0, 0};
  const int c0 = 2 * lh;

  // qk^T via WMMA, K=4, BN(channel g) applied on store
  for (int ti = 0; ti < 4; ++ti) {
    v2f aq;
    aq.x = qrows[c0 * 64 + ti * 16 + l16];
    aq.y = qrows[(c0 + 1) * 64 + ti * 16 + l16];
    for (int tj = 0; tj < 4; ++tj) {
      v2f bk;
      bk.x = krows[c0 * 64 + tj * 16 + l16];
      bk.y = krows[(c0 + 1) * 64 + tj * 16 + l16];
      v8f acc = wmma_f32(aq, bk, vz);
      int jj = tj * 16 + l16;
      if (jj < L) {
#pragma unroll
        for (int r = 0; r < 8; ++r) {
          int ii = ti * 16 + r + 8 * lh;
          if (ii < L) gsim[ii * SIMROW + jj] = acc[r] * sQK.x + sQK.y;
        }
      }
    }
  }

  // qr[i,j] = sum_c q[c,i]*rel[c][i-j+55]; kr[i,j] = sum_c k[c,j]*rel[4+c][j-i+55]
  for (int idx = lane; idx < L * L; idx += 32) {
    int i = idx / L, j = idx - i * L;
    int dq = i - j + 55, dk = j - i + 55;
    float qr = 0.f, kr = 0.f;
#pragma unroll
    for (int c = 0; c < 4; ++c) {
      qr += qrows[c * 64 + i] * lrel[c * 112 + dq];
      kr += krows[c * 64 + j] * lrel[(4 + c) * 112 + dk];
    }
    gsim[i * SIMROW + j] += (qr * sQR.x + sQR.y) + (kr * sKR.x + sKR.y);
  }

  // row softmax (wave-cooperative, shuffle reductions)
  for (int i = 0; i < L; ++i) {
    float* srow = gsim + i * SIMROW;
    float v0 = (lane < L) ? srow[lane] : -3.4e38f;
    float v1 = (lane < L - 32) ? srow[lane + 32] : -3.4e38f;
    float m = fmaxf(v0, v1);
#pragma unroll
    for (int off = 16; off; off >>= 1) m = fmaxf(m, __shfl_xor(m, off, 32));
    float e0 = (lane < L) ? __expf(v0 - m) : 0.f;
    float e1 = (lane < L - 32) ? __expf(v1 - m) : 0.f;
    float s = e0 + e1;
#pragma unroll
    for (int off = 16; off; off >>= 1) s += __shfl_xor(s, off, 32);
    float inv = 1.f / s;
    if (lane < L) srow[lane] = e0 * inv;
    if (lane < L - 32) srow[lane + 32] = e1 * inv;
  }

  // sve[c,i] = sum_j sim[i,j]*rel[8+c][i-j+55], all 32 lanes cooperating
  for (int idx = lane; idx < 8 * L; idx += 32) {
    int c = idx / L, i = idx - c * L;
    const float* er = lrel + (8 + c) * 112 + 55 + i;
    const float* sr = gsim + i * SIMROW;
    float sum = 0.f;
    for (int j = 0; j < L; ++j) sum += sr[j] * er[-j];
    gsve[c * 64 + i] = sum;
  }

  // sv[i,c] = sum_j sim[i,j]*v[c,j] via WMMA (K=56 -> 14 steps);
  // fused out-BN + (sv+sve) pair-sum + global store
  for (int ti = 0; ti < 4; ++ti) {
    v8f acc = vz;
    for (int kk = 0; kk < 14; ++kk) {
      int j0 = kk * 4 + c0;
      int i = ti * 16 + l16;
      v2f as, bv;
      bool iok = i < L;
      as.x = iok ? gsim[i * SIMROW + j0] : 0.f;
      as.y = iok ? gsim[i * SIMROW + j0 + 1] : 0.f;
      bv.x = vrows[l16 * 64 + j0];
      bv.y = vrows[l16 * 64 + j0 + 1];
      acc = wmma_f32(as, bv, acc);
    }
    int c = l16;
    if (c < 8) {
      int p = g * 8 + c;
      float2 s0 = bn_coef(bnout, 2 * CW, 2 * p);
      float2 s1 = bn_coef(bnout, 2 * CW, 2 * p + 1);
#pragma unroll
      for (int r = 0; r < 8; ++r) {
        int i = ti * 16 + r + 8 * lh;
        if (i < L) {
          float ov = (acc[r] * s0.x + s0.y) + (gsve[c * 64 + i] * s1.x + s1.y);
          if (WIDTH)
            out[(((size_t)b * CW + p) * HWD + row) * HWD + i] = ov;
          else
            out[(((size_t)b * CW + p) * HWD + i) * HWD + row] = ov;
        }
      }
    }
  }
}

extern "C" void kernel_launch(void* const* d_in, const int* in_sizes, int n_in,
                              void* d_out, int out_size, void* d_ws, size_t ws_size,
                              hipStream_t stream) {
  (void)in_sizes; (void)n_in; (void)out_size; (void)ws_size;
  const float* x        = (const float*)d_in[0];
  const float* cdw      = (const float*)d_in[1];
  const float* bn1      = (const float*)d_in[2];
  const float* h_qkvw   = (const float*)d_in[3];
  const float* h_bnqkv  = (const float*)d_in[4];
  const float* h_bnsim  = (const float*)d_in[5];
  const float* h_bnout  = (const float*)d_in[6];
  const float* h_rel    = (const float*)d_in[7];
  const float* w_qkvw   = (const float*)d_in[8];
  const float* w_bnqkv  = (const float*)d_in[9];
  const float* w_bnsim  = (const float*)d_in[10];
  const float* w_bnout  = (const float*)d_in[11];
  const float* w_rel    = (const float*)d_in[12];
  const float* cuw      = (const float*)d_in[13];
  const float* bn2      = (const float*)d_in[14];
  float* outp = (float*)d_out;

  float* y1 = (float*)d_ws;                         // (16,64,56,56)
  float* y2 = y1 + (size_t)BSZ * CW * NPOS;
  float* y3 = y2 + (size_t)BSZ * CW * NPOS;

  // allow >64KB dynamic LDS (CDNA5 WGP has 320KB)
  hipFuncSetAttribute(reinterpret_cast<const void*>(&attn_kernel<false>),
                      hipFuncAttributeMaxDynamicSharedMemorySize, SMEM_BYTES);
  hipFuncSetAttribute(reinterpret_cast<const void*>(&attn_kernel<true>),
                      hipFuncAttributeMaxDynamicSharedMemorySize, SMEM_BYTES);

  conv_down_kernel<<<392, 256, 0, stream>>>(x, cdw, bn1, y1);
  attn_kernel<false><<<BSZ * HWD, 256, SMEM_BYTES, stream>>>(
      y1, h_qkvw, h_bnqkv, h_bnsim, h_bnout, h_rel, y2);
  attn_kernel<true><<<BSZ * HWD, 256, SMEM_BYTES, stream>>>(
      y2, w_qkvw, w_bnqkv, w_bnsim, w_bnout, w_rel, y3);
  conv_up_kernel<<<392, 256, 0, stream>>>(y3, x, cuw, bn2, outp);
}